// BalancedBatchNorm2d_33285996544682
// MI455X (gfx1250) — compile-verified
//
#include <hip/hip_runtime.h>
#include <hip/hip_bf16.h>

#define NUM_CLASSES 100
#define CCH 64          // channels
#define BB  512         // batch
#define HW  1024        // 32*32
#define NSLICE (BB * CCH)

// workspace layout (floats)
#define WS_SUMX  0      // [64]  sum X per channel
#define WS_SUMX2 64     // [64]  sum X^2 per channel
#define WS_WSUM  128    // [64]  sum over b of w[b]*sliceSum(b,c)
#define WS_W     192    // [512] per-sample weight 1/cnt[label[b]]
#define WS_SCALE 704    // [64]
#define WS_SHIFT 768    // [64]

typedef __attribute__((ext_vector_type(2))) float v2f;
typedef __attribute__((ext_vector_type(4))) float v4f;
typedef __attribute__((ext_vector_type(8))) float v8f;

// ---------- phase 0: class counts -> per-sample weights; zero accumulators ----
__global__ __launch_bounds__(512) void bb_phase0(const long long* __restrict__ label,
                                                 float* __restrict__ ws) {
    __shared__ float scnt[NUM_CLASSES];
    const int t = threadIdx.x;                 // 512 threads == batch size
    if (t < NUM_CLASSES) scnt[t] = 0.0f;
    if (t < 192) ws[t] = 0.0f;                 // zero sumX / sumX2 / wSum each call
    __syncthreads();
    const int lbl = (int)label[t];
    atomicAdd(&scnt[lbl], 1.0f);
    __syncthreads();
    ws[WS_W + t] = 1.0f / scnt[lbl];
}

// ---------- phase 1: per-(b,c) slice sums via f32 WMMA, fold into channels ----
__global__ __launch_bounds__(256) void bb_reduce(const float* __restrict__ X,
                                                 float* __restrict__ ws) {
    const int lane  = threadIdx.x & 31;
    const int slice = blockIdx.x * 8 + (threadIdx.x >> 5);   // one wave = one slice
    const float* p = X + (size_t)slice * HW;

    v2f ones; ones[0] = 1.0f; ones[1] = 1.0f;                // A = 16x4 all-ones
    v8f acc1 = {};                                           // accumulates sum(x)
    v8f acc2 = {};                                           // accumulates sum(x^2)

#pragma unroll
    for (int it = 0; it < 16; ++it) {                        // 16 * 64 = 1024 elems
        const v2f x = *(const v2f*)(p + it * 64 + lane * 2);
        v2f bq; bq[0] = x[0] * x[0]; bq[1] = x[1] * x[1];
        // D = ones(16x4) * B(4x16) + C : every D row = column sums of the tile
        acc1 = __builtin_amdgcn_wmma_f32_16x16x4_f32(false, ones, false, x,
                                                     (short)0, acc1, false, false);
        acc2 = __builtin_amdgcn_wmma_f32_16x16x4_f32(false, ones, false, bq,
                                                     (short)0, acc2, false, false);
    }

    // VGPR0 over 32 lanes holds two identical rows of D -> sum == 2 * total
    float s1 = acc1[0];
    float s2 = acc2[0];
#pragma unroll
    for (int off = 16; off >= 1; off >>= 1) {
        s1 += __shfl_xor(s1, off, 32);
        s2 += __shfl_xor(s2, off, 32);
    }
    s1 *= 0.5f;
    s2 *= 0.5f;

    if (lane == 0) {
        const int b = slice >> 6;
        const int c = slice & 63;
        const float w = ws[WS_W + b];
        atomicAdd(ws + WS_SUMX  + c, s1);
        atomicAdd(ws + WS_SUMX2 + c, s2);
        atomicAdd(ws + WS_WSUM  + c, w * s1);
    }
}

// ---------- phase 2: per-channel mean/var -> scale/shift --------------------
__global__ __launch_bounds__(64) void bb_finalize(float* __restrict__ ws,
                                                  const float* __restrict__ gamma,
                                                  const float* __restrict__ beta) {
    const int c = threadIdx.x;                 // 64 threads
    const float invLHW = 1.0f / ((float)NUM_CLASSES * (float)HW);
    const float invN   = 1.0f / ((float)BB * (float)HW);
    const float mean = ws[WS_WSUM + c] * invLHW;
    const float ex   = ws[WS_SUMX  + c] * invN;
    const float ex2  = ws[WS_SUMX2 + c] * invN;
    const float var  = ex2 - 2.0f * mean * ex + mean * mean;
    const float sc   = gamma[c] * rsqrtf(var + 1e-6f);
    ws[WS_SCALE + c] = sc;
    ws[WS_SHIFT + c] = beta[c] - mean * sc;
}

// ---------- phase 3: Y = scale[c]*X + shift[c]; NT stores keep X hot in L2 --
__global__ __launch_bounds__(256) void bb_normalize(const v4f* __restrict__ X4,
                                                    const float* __restrict__ ws,
                                                    v4f* __restrict__ Y4) {
    const int c = blockIdx.x & 63;             // one block == one (b,c) slice
    const float sc = ws[WS_SCALE + c];
    const float sh = ws[WS_SHIFT + c];
    const size_t i = (size_t)blockIdx.x * 256 + threadIdx.x;
    const v4f x = X4[i];
    v4f y;
    y[0] = fmaf(sc, x[0], sh);
    y[1] = fmaf(sc, x[1], sh);
    y[2] = fmaf(sc, x[2], sh);
    y[3] = fmaf(sc, x[3], sh);
    __builtin_nontemporal_store(y, &Y4[i]);
}

extern "C" void kernel_launch(void* const* d_in, const int* in_sizes, int n_in,
                              void* d_out, int out_size, void* d_ws, size_t ws_size,
                              hipStream_t stream) {
    const float*     X     = (const float*)d_in[0];
    const long long* label = (const long long*)d_in[1];
    const float*     gamma = (const float*)d_in[2];
    const float*     beta  = (const float*)d_in[3];
    float* ws  = (float*)d_ws;
    float* out = (float*)d_out;

    bb_phase0<<<1, 512, 0, stream>>>(label, ws);
    bb_reduce<<<NSLICE / 8, 256, 0, stream>>>(X, ws);        // 4096 blocks, 8 waves each
    bb_finalize<<<1, 64, 0, stream>>>(ws, gamma, beta);
    bb_normalize<<<NSLICE, 256, 0, stream>>>((const v4f*)X, ws, (v4f*)out);
}